// Seq_Classification_3470333575208
// MI455X (gfx1250) — compile-verified
//
#include <hip/hip_runtime.h>
#include <hip/hip_bf16.h>

#define N_NODES  100000
#define N_EDGES  600000
#define N_GRAPHS 64

typedef __attribute__((ext_vector_type(2))) float v2f;
typedef __attribute__((ext_vector_type(8))) float v8f;

// V_WMMA_F32_16X16X4_F32: D(16x16 f32) = A(16x4 f32) * B(4x16 f32) + C
// 8-arg pattern: (neg_a, A, neg_b, B, c_mod, C, reuse_a, reuse_b)
__device__ __forceinline__ v8f wmma4(v2f a, v2f b, v8f c) {
  return __builtin_amdgcn_wmma_f32_16x16x4_f32(
      false, a, false, b, (short)0, c, false, false);
}

// ---------------------------------------------------------------------------
// Edge scatter, layer 1: acc1[dst] += x[src] (128 feats), cnt[dst] += 1
// One wave per edge; each lane handles 4 consecutive features (float4 load).
// ---------------------------------------------------------------------------
__global__ void scatter1_kernel(const float* __restrict__ x,
                                const int* __restrict__ src,
                                const int* __restrict__ dst,
                                float* __restrict__ acc,
                                float* __restrict__ cnt) {
  const int tid  = blockIdx.x * blockDim.x + threadIdx.x;
  const int e    = tid >> 5;
  const int lane = tid & 31;
  if (e >= N_EDGES) return;
  const int s = src[e];
  const int d = dst[e];
  const float4 v = *(const float4*)(x + (size_t)s * 128 + lane * 4);
  float* p = acc + (size_t)d * 128 + lane * 4;
  atomicAdd(p + 0, v.x);
  atomicAdd(p + 1, v.y);
  atomicAdd(p + 2, v.z);
  atomicAdd(p + 3, v.w);
  if (lane == 0) atomicAdd(cnt + d, 1.0f);
}

// ---------------------------------------------------------------------------
// Edge scatter, layer 2: acc2[dst] += h1[src] (64 feats)
// 16 lanes per edge, 4 feats each (float4).
// ---------------------------------------------------------------------------
__global__ void scatter2_kernel(const float* __restrict__ h,
                                const int* __restrict__ src,
                                const int* __restrict__ dst,
                                float* __restrict__ acc) {
  const int tid  = blockIdx.x * blockDim.x + threadIdx.x;
  const int e    = tid >> 4;
  const int lane = tid & 15;
  if (e >= N_EDGES) return;
  const int s = src[e];
  const int d = dst[e];
  const float4 v = *(const float4*)(h + (size_t)s * 64 + lane * 4);
  float* p = acc + (size_t)d * 64 + lane * 4;
  atomicAdd(p + 0, v.x);
  atomicAdd(p + 1, v.y);
  atomicAdd(p + 2, v.z);
  atomicAdd(p + 3, v.w);
}

// ---------------------------------------------------------------------------
// Layer-1 node GEMM: h1 = relu((acc1/cnt) @ W1l^T + x @ W1r^T + b1)
// One wave per 16x16 output tile; K = 128 for each of the two terms.
// ---------------------------------------------------------------------------
__global__ void gemm1_kernel(const float* __restrict__ acc,
                             const float* __restrict__ cnt,
                             const float* __restrict__ x,
                             const float* __restrict__ Wl,
                             const float* __restrict__ Wr,
                             const float* __restrict__ bias,
                             float* __restrict__ h1) {
  const int lane = threadIdx.x & 31;
  const int wave = (blockIdx.x * blockDim.x + threadIdx.x) >> 5;
  const int mt = wave >> 2;   // 6250 row tiles (100000/16 exactly)
  const int ct = wave & 3;    // 4 column tiles (64/16)
  if (mt >= N_NODES / 16) return;
  const int m    = lane & 15; // A row / B col / D col within tile
  const int half = lane >> 4;
  const int row0 = mt * 16;
  const int col0 = ct * 16;

  const float c    = cnt[row0 + m];
  const float rinv = 1.0f / fmaxf(c, 1.0f);

  const float* arow = acc + (size_t)(row0 + m) * 128 + half * 2;
  const float* xrow = x   + (size_t)(row0 + m) * 128 + half * 2;
  const float* blp  = Wl  + (size_t)(col0 + m) * 128 + half * 2;
  const float* brp  = Wr  + (size_t)(col0 + m) * 128 + half * 2;

  v8f d = {};
#pragma unroll 8
  for (int k = 0; k < 128; k += 4) {
    v2f a = *(const v2f*)(arow + k);
    a.x *= rinv; a.y *= rinv;
    v2f b = *(const v2f*)(blp + k);
    d = wmma4(a, b, d);
  }
#pragma unroll 8
  for (int k = 0; k < 128; k += 4) {
    v2f a = *(const v2f*)(xrow + k);
    v2f b = *(const v2f*)(brp + k);
    d = wmma4(a, b, d);
  }

  const int   col = col0 + m;
  const float bv  = bias[col];
  float* outp = h1 + (size_t)(row0 + half * 8) * 64 + col;
#pragma unroll
  for (int v = 0; v < 8; ++v)
    outp[(size_t)v * 64] = fmaxf(d[v] + bv, 0.0f);
}

// ---------------------------------------------------------------------------
// Layer-2 node GEMM: h2 = (acc2/cnt) @ W2l^T + h1 @ W2r^T + b2   (N=16, K=64)
// ---------------------------------------------------------------------------
__global__ void gemm2_kernel(const float* __restrict__ acc,
                             const float* __restrict__ cnt,
                             const float* __restrict__ h1,
                             const float* __restrict__ Wl,
                             const float* __restrict__ Wr,
                             const float* __restrict__ bias,
                             float* __restrict__ h2) {
  const int lane = threadIdx.x & 31;
  const int mt   = (blockIdx.x * blockDim.x + threadIdx.x) >> 5;
  if (mt >= N_NODES / 16) return;
  const int m    = lane & 15;
  const int half = lane >> 4;
  const int row0 = mt * 16;

  const float c    = cnt[row0 + m];
  const float rinv = 1.0f / fmaxf(c, 1.0f);

  const float* arow = acc + (size_t)(row0 + m) * 64 + half * 2;
  const float* hrow = h1  + (size_t)(row0 + m) * 64 + half * 2;
  const float* blp  = Wl  + (size_t)m * 64 + half * 2;
  const float* brp  = Wr  + (size_t)m * 64 + half * 2;

  v8f d = {};
#pragma unroll 8
  for (int k = 0; k < 64; k += 4) {
    v2f a = *(const v2f*)(arow + k);
    a.x *= rinv; a.y *= rinv;
    v2f b = *(const v2f*)(blp + k);
    d = wmma4(a, b, d);
  }
#pragma unroll 8
  for (int k = 0; k < 64; k += 4) {
    v2f a = *(const v2f*)(hrow + k);
    v2f b = *(const v2f*)(brp + k);
    d = wmma4(a, b, d);
  }

  const float bv = bias[m];
  float* outp = h2 + (size_t)(row0 + half * 8) * 16 + m;
#pragma unroll
  for (int v = 0; v < 8; ++v)
    outp[(size_t)v * 16] = d[v] + bv;
}

// ---------------------------------------------------------------------------
// Graph pooling: LDS-staged segment sum of h2 by batch_ids + node counts.
// ---------------------------------------------------------------------------
__global__ void pool_kernel(const float* __restrict__ h2,
                            const int* __restrict__ bids,
                            float* __restrict__ gsum,
                            float* __restrict__ gcnt) {
  __shared__ float lsum[N_GRAPHS * 16];
  __shared__ float lcnt[N_GRAPHS];
  for (int i = threadIdx.x; i < N_GRAPHS * 16; i += blockDim.x) lsum[i] = 0.0f;
  for (int i = threadIdx.x; i < N_GRAPHS; i += blockDim.x) lcnt[i] = 0.0f;
  __syncthreads();
  const int total  = N_NODES * 16;
  const int stride = gridDim.x * blockDim.x;
  for (int i = blockIdx.x * blockDim.x + threadIdx.x; i < total; i += stride) {
    const int r = i >> 4;
    const int c = i & 15;
    const int g = bids[r];
    atomicAdd(&lsum[g * 16 + c], h2[i]);
    if (c == 0) atomicAdd(&lcnt[g], 1.0f);
  }
  __syncthreads();
  for (int i = threadIdx.x; i < N_GRAPHS * 16; i += blockDim.x)
    atomicAdd(&gsum[i], lsum[i]);
  for (int i = threadIdx.x; i < N_GRAPHS; i += blockDim.x)
    atomicAdd(&gcnt[i], lcnt[i]);
}

// ---------------------------------------------------------------------------
// Adapter head: out[:, :64] = bert_cls @ W_adapt^T + b_adapt  (64x64x768)
// 16 waves, one 16x16 tile each.
// ---------------------------------------------------------------------------
__global__ void adapt_kernel(const float* __restrict__ bert,
                             const float* __restrict__ W,
                             const float* __restrict__ bias,
                             float* __restrict__ out) {
  const int lane = threadIdx.x & 31;
  const int wave = (blockIdx.x * blockDim.x + threadIdx.x) >> 5;
  if (wave >= 16) return;
  const int mt = wave >> 2;
  const int ct = wave & 3;
  const int m    = lane & 15;
  const int half = lane >> 4;
  const int row0 = mt * 16;
  const int col0 = ct * 16;

  const float* arow = bert + (size_t)(row0 + m) * 768 + half * 2;
  const float* brow = W    + (size_t)(col0 + m) * 768 + half * 2;

  v8f d = {};
#pragma unroll 8
  for (int k = 0; k < 768; k += 4) {
    v2f a = *(const v2f*)(arow + k);
    v2f b = *(const v2f*)(brow + k);
    d = wmma4(a, b, d);
  }

  const int   col = col0 + m;
  const float bv  = bias[col];
  float* outp = out + (size_t)(row0 + half * 8) * 80 + col;
#pragma unroll
  for (int v = 0; v < 8; ++v)
    outp[(size_t)v * 80] = d[v] + bv;
}

// ---------------------------------------------------------------------------
// Final head: out[:, 64:80] = (gsum/gcnt) @ Wm^T + bm    (64x16x16)
// 4 waves in one block.
// ---------------------------------------------------------------------------
__global__ void final_kernel(const float* __restrict__ gsum,
                             const float* __restrict__ gcnt,
                             const float* __restrict__ Wm,
                             const float* __restrict__ bm,
                             float* __restrict__ out) {
  const int lane = threadIdx.x & 31;
  const int wave = threadIdx.x >> 5;
  if (wave >= 4) return;
  const int m    = lane & 15;
  const int half = lane >> 4;
  const int row0 = wave * 16;

  const float c    = gcnt[row0 + m];
  const float rinv = 1.0f / fmaxf(c, 1.0f);

  const float* arow = gsum + (size_t)(row0 + m) * 16 + half * 2;
  const float* brow = Wm   + (size_t)m * 16 + half * 2;

  v8f d = {};
#pragma unroll
  for (int k = 0; k < 16; k += 4) {
    v2f a = *(const v2f*)(arow + k);
    a.x *= rinv; a.y *= rinv;
    v2f b = *(const v2f*)(brow + k);
    d = wmma4(a, b, d);
  }

  const float bv = bm[m];
  float* outp = out + (size_t)(row0 + half * 8) * 80 + 64 + m;
#pragma unroll
  for (int v = 0; v < 8; ++v)
    outp[(size_t)v * 80] = d[v] + bv;
}

// ---------------------------------------------------------------------------
// Workspace layout (bytes):
//   [0, 51.2M)        acc1 (100000*128 f32); reused as acc2 (100000*64) later
//   [51.2M, 76.8M)    h1   (100000*64 f32)
//   [76.8M, 83.2M)    h2   (100000*16 f32)
//   [83.2M, 83.6M)    cnt  (100000 f32)
//   [83.6M, +4KB)     gsum (64*16 f32)
//   [+4KB, +256B)     gcnt (64 f32)
// ---------------------------------------------------------------------------
extern "C" void kernel_launch(void* const* d_in, const int* in_sizes, int n_in,
                              void* d_out, int out_size, void* d_ws, size_t ws_size,
                              hipStream_t stream) {
  const float* bert    = (const float*)d_in[0];
  const float* node_x  = (const float*)d_in[1];
  const int*   edges   = (const int*)d_in[2];
  const int*   bids    = (const int*)d_in[3];
  const float* W_adapt = (const float*)d_in[4];
  const float* b_adapt = (const float*)d_in[5];
  const float* W1l     = (const float*)d_in[6];
  const float* W1r     = (const float*)d_in[7];
  const float* b1      = (const float*)d_in[8];
  const float* W2l     = (const float*)d_in[9];
  const float* W2r     = (const float*)d_in[10];
  const float* b2      = (const float*)d_in[11];
  const float* Wm      = (const float*)d_in[12];
  const float* bm      = (const float*)d_in[13];
  float* out = (float*)d_out;

  const int* src = edges;
  const int* dst = edges + N_EDGES;

  char* ws = (char*)d_ws;
  float* acc1 = (float*)(ws);
  float* h1   = (float*)(ws + 51200000);
  float* h2   = (float*)(ws + 76800000);
  float* cnt  = (float*)(ws + 83200000);
  float* gsum = (float*)(ws + 83600000);
  float* gcnt = (float*)(ws + 83604096);
  float* acc2 = acc1;  // acc1 is dead after gemm1

  // Zero acc1/cnt/gsum/gcnt (h1/h2 fully overwritten).
  hipMemsetAsync(d_ws, 0, 83604352, stream);

  // Independent adapter head -> out[:, :64]
  adapt_kernel<<<2, 256, 0, stream>>>(bert, W_adapt, b_adapt, out);

  // Layer 1: scatter + WMMA GEMM
  scatter1_kernel<<<(N_EDGES * 32) / 256, 256, 0, stream>>>(node_x, src, dst, acc1, cnt);
  gemm1_kernel<<<(N_NODES / 16 * 4 * 32) / 256, 256, 0, stream>>>(
      acc1, cnt, node_x, W1l, W1r, b1, h1);

  // Layer 2: re-zero aliased accumulator, scatter + WMMA GEMM
  hipMemsetAsync(acc2, 0, (size_t)N_NODES * 64 * sizeof(float), stream);
  scatter2_kernel<<<(N_EDGES * 16) / 256, 256, 0, stream>>>(h1, src, dst, acc2);
  gemm2_kernel<<<(N_NODES / 16 * 32 + 255) / 256, 256, 0, stream>>>(
      acc2, cnt, h1, W2l, W2r, b2, h2);

  // Graph mean-pool + final linear -> out[:, 64:80]
  pool_kernel<<<128, 256, 0, stream>>>(h2, bids, gsum, gcnt);
  final_kernel<<<1, 128, 0, stream>>>(gsum, gcnt, Wm, bm, out);
}